// BiLSTM_CRF_82463372083238
// MI455X (gfx1250) — compile-verified
//
#include <hip/hip_runtime.h>
#include <hip/hip_bf16.h>

typedef __attribute__((ext_vector_type(2))) float v2f;
typedef __attribute__((ext_vector_type(8))) float v8f;

#define S_LEN   4096
#define EMBD    256
#define HIDD    128     // per-direction hidden
#define G4H     512     // 4*H
#define TT      12
#define START_TAG 10
#define END_TAG   11
#define FSTRIDE   16    // padded feats row stride

__device__ __forceinline__ float sigmoidf_(float x) {
    return 1.0f / (1.0f + __expf(-x));
}

// ---------------------------------------------------------------------------
// Kernel 1: G[d][t][0:512] = W_ih_d @ emb[sentence[t]] + (b_ih_d + b_hh_d)
// WMMA f32 16x16x4; embedding gather fused via LDS strip.
// grid = (S/16, 512/128, 2dirs), block = 256 (8 waves, each one 16x16 N-tile)
// ---------------------------------------------------------------------------
__global__ __launch_bounds__(256) void bilstm_gates_wmma(
    const int*   __restrict__ sent, const float* __restrict__ emb,
    const float* __restrict__ Wf, const float* __restrict__ bif, const float* __restrict__ bhf,
    const float* __restrict__ Wb, const float* __restrict__ bib, const float* __restrict__ bhb,
    float* __restrict__ Gf, float* __restrict__ Gb)
{
    __shared__ float Xs[16][260];                    // 16 rows x 256 cols, padded vs bank conflicts
    const int t0  = blockIdx.x * 16;
    const int dir = blockIdx.z;
    const float* W  = dir ? Wb  : Wf;
    const float* bi = dir ? bib : bif;
    const float* bh = dir ? bhb : bhf;
    float* G = dir ? Gb : Gf;

    {   // cooperative gather: 16 threads per row, 16 floats each
        const int tid = threadIdx.x;
        const int m   = tid >> 4;
        const int cb  = (tid & 15) << 4;
        const float* src = emb + (size_t)sent[t0 + m] * EMBD + cb;
        float* dst = &Xs[m][cb];
        #pragma unroll
        for (int j = 0; j < 16; j += 4)
            *(float4*)(dst + j) = *(const float4*)(src + j);
    }
    __syncthreads();

    const int lane = threadIdx.x & 31;
    const int wv   = threadIdx.x >> 5;
    const int row  = lane & 15;                      // M for A, N for B
    const int kh   = lane >> 4;                      // K half select
    const int n0   = blockIdx.y * 128 + wv * 16;
    const float* wrow = W + (size_t)(n0 + row) * EMBD;

    v8f acc = {0.f,0.f,0.f,0.f,0.f,0.f,0.f,0.f};
    #pragma unroll 4
    for (int k0 = 0; k0 < EMBD; k0 += 4) {
        const int k = k0 + 2 * kh;
        v2f a = *(const v2f*)&Xs[row][k];            // A: x rows (time-major)
        v2f b = *(const v2f*)(wrow + k);             // B: W_ih^T chunk
        acc = __builtin_amdgcn_wmma_f32_16x16x4_f32(
                  false, a, false, b, (short)0, acc, false, false);
    }

    const float bs = bi[n0 + row] + bh[n0 + row];
    #pragma unroll
    for (int r = 0; r < 8; r++)
        G[(size_t)(t0 + r + 8 * kh) * G4H + n0 + row] = acc[r] + bs;
}

// ---------------------------------------------------------------------------
// Kernel 2: LSTM recurrence. grid = 2 (one block per direction), block = 512.
// Each thread owns one Whh row (128 f32) in registers; h broadcast via LDS.
// Backward direction consumes G_b[S-1-s] and writes h_b[S-1-s].
// ---------------------------------------------------------------------------
__global__ __launch_bounds__(512) void bilstm_recurrence(
    const float* __restrict__ Gf, const float* __restrict__ Gb,
    const float* __restrict__ Whf, const float* __restrict__ Whb,
    const float* __restrict__ h0, const float* __restrict__ c0,
    float* __restrict__ hf, float* __restrict__ hb)
{
    __shared__ float h_sh[HIDD];
    __shared__ float pre[G4H];
    const int d = blockIdx.x;
    const int r = threadIdx.x;
    const float* Whh = d ? Whb : Whf;
    const float* G   = d ? Gb  : Gf;
    float* hout      = d ? hb  : hf;

    float w[HIDD];
    #pragma unroll
    for (int k = 0; k < HIDD; k++) w[k] = Whh[(size_t)r * HIDD + k];

    float c = 0.f;
    if (r < HIDD) { h_sh[r] = h0[d * HIDD + r]; c = c0[d * HIDD + r]; }
    __syncthreads();

    for (int s = 0; s < S_LEN; s++) {
        const int t = d ? (S_LEN - 1 - s) : s;
        float acc = G[(size_t)t * G4H + r];
        #pragma unroll
        for (int k = 0; k < HIDD; k++) acc = fmaf(w[k], h_sh[k], acc);
        pre[r] = acc;
        __syncthreads();
        if (r < HIDD) {
            const float ig = sigmoidf_(pre[r]);
            const float fg = sigmoidf_(pre[HIDD + r]);
            const float gg = tanhf(pre[2 * HIDD + r]);
            const float og = sigmoidf_(pre[3 * HIDD + r]);
            c = fg * c + ig * gg;
            const float h = og * tanhf(c);
            h_sh[r] = h;
            hout[(size_t)t * HIDD + r] = h;
        }
        __syncthreads();
    }
}

// ---------------------------------------------------------------------------
// Kernel 3: feats = [h_f|h_b] @ W_tag^T + b_tag, T=12 padded to 16 cols.
// grid = S/16 waves, block = 32 (one wave, one 16x16 tile).
// ---------------------------------------------------------------------------
__global__ __launch_bounds__(32) void tag_feats_wmma(
    const float* __restrict__ hf, const float* __restrict__ hb,
    const float* __restrict__ Wtag, const float* __restrict__ btag,
    float* __restrict__ feats)
{
    const int t0   = blockIdx.x * 16;
    const int lane = threadIdx.x & 31;
    const int row  = lane & 15;
    const int kh   = lane >> 4;

    const float* hfr = hf + (size_t)(t0 + row) * HIDD;
    const float* hbr = hb + (size_t)(t0 + row) * HIDD;
    const float* wtr = Wtag + (size_t)row * EMBD;   // only valid for row < TT

    v8f acc = {0.f,0.f,0.f,0.f,0.f,0.f,0.f,0.f};
    #pragma unroll 4
    for (int k0 = 0; k0 < EMBD; k0 += 4) {
        const int k = k0 + 2 * kh;                  // even; never straddles the 128 split
        v2f a = (k < HIDD) ? *(const v2f*)(hfr + k)
                           : *(const v2f*)(hbr + (k - HIDD));
        v2f b;
        if (row < TT) b = *(const v2f*)(wtr + k);
        else { b.x = 0.f; b.y = 0.f; }              // zero-padded tag columns 12..15
        acc = __builtin_amdgcn_wmma_f32_16x16x4_f32(
                  false, a, false, b, (short)0, acc, false, false);
    }
    const float bt = (row < TT) ? btag[row] : 0.f;
    #pragma unroll
    for (int r = 0; r < 8; r++)
        feats[(size_t)(t0 + r + 8 * kh) * FSTRIDE + row] = acc[r] + bt;
}

// ---------------------------------------------------------------------------
// Kernel 4: CRF forward scan + gold score + final NLL. 1 block, 1 wave.
// ---------------------------------------------------------------------------
__global__ __launch_bounds__(32) void crf_nll(
    const float* __restrict__ feats, const float* __restrict__ trans,
    const int* __restrict__ gold, float* __restrict__ out)
{
    __shared__ float alpha[TT];
    __shared__ float red[32];
    const int lane = threadIdx.x;

    float tr[TT];
    if (lane < TT) {
        #pragma unroll
        for (int k = 0; k < TT; k++) tr[k] = trans[lane * TT + k];  // trans[next=lane][prev=k]
        alpha[lane] = (lane == START_TAG) ? 0.f : -1000000.0f;
    }
    __syncthreads();

    for (int t = 0; t < S_LEN; t++) {
        float nv = 0.f;
        if (lane < TT) {
            float m = -3.4e38f;
            #pragma unroll
            for (int k = 0; k < TT; k++) m = fmaxf(m, alpha[k] + tr[k]);
            float s = 0.f;
            #pragma unroll
            for (int k = 0; k < TT; k++) s += __expf(alpha[k] + tr[k] - m);
            nv = m + __logf(s) + feats[t * FSTRIDE + lane];
        }
        __syncthreads();
        if (lane < TT) alpha[lane] = nv;
        __syncthreads();
    }

    // gold path score: per-lane partials over strided timesteps
    float part = 0.f;
    for (int t = lane; t < S_LEN; t += 32) {
        const int cur  = gold[t];
        const int prev = (t == 0) ? START_TAG : gold[t - 1];
        part += trans[cur * TT + prev] + feats[t * FSTRIDE + cur];
    }
    red[lane] = part;
    __syncthreads();

    if (lane == 0) {
        float gsum = 0.f;
        for (int k = 0; k < 32; k++) gsum += red[k];
        gsum += trans[END_TAG * TT + gold[S_LEN - 1]];

        float m = -3.4e38f;
        for (int k = 0; k < TT; k++) m = fmaxf(m, alpha[k] + trans[END_TAG * TT + k]);
        float s = 0.f;
        for (int k = 0; k < TT; k++) s += __expf(alpha[k] + trans[END_TAG * TT + k] - m);
        const float fwd = m + __logf(s);
        out[0] = fwd - gsum;
    }
}

// ---------------------------------------------------------------------------
extern "C" void kernel_launch(void* const* d_in, const int* in_sizes, int n_in,
                              void* d_out, int out_size, void* d_ws, size_t ws_size,
                              hipStream_t stream) {
    const int*   sentence = (const int*)  d_in[0];
    const int*   gold     = (const int*)  d_in[1];
    const float* emb      = (const float*)d_in[2];
    const float* W_ih_f   = (const float*)d_in[3];
    const float* W_hh_f   = (const float*)d_in[4];
    const float* b_ih_f   = (const float*)d_in[5];
    const float* b_hh_f   = (const float*)d_in[6];
    const float* W_ih_b   = (const float*)d_in[7];
    const float* W_hh_b   = (const float*)d_in[8];
    const float* b_ih_b   = (const float*)d_in[9];
    const float* b_hh_b   = (const float*)d_in[10];
    const float* h0       = (const float*)d_in[11];
    const float* c0       = (const float*)d_in[12];
    const float* W_tag    = (const float*)d_in[13];
    const float* b_tag    = (const float*)d_in[14];
    const float* trans    = (const float*)d_in[15];

    float* ws    = (float*)d_ws;
    float* G_f   = ws;                               // [S,512]
    float* G_b   = G_f + (size_t)S_LEN * G4H;        // [S,512]
    float* h_f   = G_b + (size_t)S_LEN * G4H;        // [S,128]
    float* h_b   = h_f + (size_t)S_LEN * HIDD;       // [S,128]
    float* feats = h_b + (size_t)S_LEN * HIDD;       // [S,16] padded

    bilstm_gates_wmma<<<dim3(S_LEN / 16, 4, 2), 256, 0, stream>>>(
        sentence, emb, W_ih_f, b_ih_f, b_hh_f, W_ih_b, b_ih_b, b_hh_b, G_f, G_b);

    bilstm_recurrence<<<2, 512, 0, stream>>>(
        G_f, G_b, W_hh_f, W_hh_b, h0, c0, h_f, h_b);

    tag_feats_wmma<<<S_LEN / 16, 32, 0, stream>>>(
        h_f, h_b, W_tag, b_tag, feats);

    crf_nll<<<1, 32, 0, stream>>>(feats, trans, gold, (float*)d_out);
}